// Net_81140522156519
// MI455X (gfx1250) — compile-verified
//
#include <hip/hip_runtime.h>
#include <math.h>

#define NN      32768
#define EE      327680
#define NCH     16          // padded channel width for all layers
#define NBINS   125         // K^3
#define MCAP    (EE * 8 + NBINS * 16)   // padded (edge,corner) pair capacity
#define NTILES  (MCAP / 16)             // 163965 exactly

typedef __attribute__((ext_vector_type(2))) float v2f;
typedef __attribute__((ext_vector_type(8))) float v8f;

// ---------------- spline basis helper (degree-1, K=5, DIM=3) ----------------
__device__ __forceinline__ void spline_corner(float p0, float p1, float p2, int s,
                                              float& bs, int& g) {
    float v0 = p0 * 4.0f, v1 = p1 * 4.0f, v2 = p2 * 4.0f;
    float l0 = fminf(fmaxf(floorf(v0), 0.0f), 3.0f);
    float l1 = fminf(fmaxf(floorf(v1), 0.0f), 3.0f);
    float l2 = fminf(fmaxf(floorf(v2), 0.0f), 3.0f);
    float f0 = v0 - l0, f1 = v1 - l1, f2 = v2 - l2;
    int b0 = s & 1, b1 = (s >> 1) & 1, b2 = (s >> 2) & 1;
    float w0 = b0 ? f0 : 1.0f - f0;
    float w1 = b1 ? f1 : 1.0f - f1;
    float w2 = b2 ? f2 : 1.0f - f2;
    bs = w0 * w1 * w2;
    int i0 = (int)l0 + b0, i1 = (int)l1 + b1, i2 = (int)l2 + b2;
    g = i0 + 5 * i1 + 25 * i2;
}

// ---------------- init: zero pair arrays, h0, reduction scalars, counts -----
__global__ void k_init(float* red, int* counts, float* h0,
                       float2* pairSB, int* pairDst) {
    int i = blockIdx.x * blockDim.x + threadIdx.x;
    if (i < MCAP) { pairSB[i] = make_float2(0.0f, 0.0f); pairDst[i] = 0; }
    if (i < NN * NCH) h0[i] = 0.0f;
    if (i < NBINS) counts[i] = 0;
    if (i < 2) red[i] = 0.0f;
}

// ---------------- mean / var reduction for normalize() ---------------------
__global__ void k_reduce(const float* __restrict__ x, float* red) {
    __shared__ float s1[256], s2[256];
    int tid = threadIdx.x;
    int i = blockIdx.x * 256 + tid;
    float v = (i < NN) ? x[i] : 0.0f;
    s1[tid] = v; s2[tid] = v * v;
    __syncthreads();
    for (int s = 128; s > 0; s >>= 1) {
        if (tid < s) { s1[tid] += s1[tid + s]; s2[tid] += s2[tid + s]; }
        __syncthreads();
    }
    if (tid == 0) { atomicAdd(&red[0], s1[0]); atomicAdd(&red[1], s2[0]); }
}

// h0[:,0] = x - mean/std(ddof=1); other 15 cols already zero
__global__ void k_normalize(const float* __restrict__ x, const float* __restrict__ red,
                            float* __restrict__ h0) {
    int n = blockIdx.x * blockDim.x + threadIdx.x;
    if (n >= NN) return;
    float sum = red[0], sumsq = red[1];
    float mean = sum / (float)NN;
    float var = (sumsq - sum * sum / (float)NN) / (float)(NN - 1);
    float c = mean / sqrtf(var);
    h0[n * NCH] = x[n] - c;
}

// ---------------- bin histogram over (edge, corner) pairs -------------------
__global__ void k_count(const float* __restrict__ pseudo, int* counts) {
    int e = blockIdx.x * blockDim.x + threadIdx.x;
    if (e >= EE) return;
    float p0 = pseudo[e * 3 + 0], p1 = pseudo[e * 3 + 1], p2 = pseudo[e * 3 + 2];
#pragma unroll
    for (int s = 0; s < 8; ++s) {
        float bs; int g;
        spline_corner(p0, p1, p2, s, bs, g);
        atomicAdd(&counts[g], 1);
    }
}

// exclusive scan with per-bin pad-to-16 (125 bins: serial on one lane is fine)
__global__ void k_scan(const int* __restrict__ counts, int* offsets, int* cursors) {
    if (threadIdx.x == 0 && blockIdx.x == 0) {
        int off = 0;
        for (int g = 0; g < NBINS; ++g) {
            offsets[g] = off;
            cursors[g] = off;
            off += (counts[g] + 15) & ~15;
        }
        offsets[NBINS] = off;
    }
}

// tile -> bin map, computed once (tiles never straddle bins: bins padded to 16)
__global__ void k_tileg(const int* __restrict__ offsets, int* __restrict__ tileG) {
    int t = blockIdx.x * blockDim.x + threadIdx.x;
    if (t >= NTILES) return;
    int off = t << 4;
    int g = NBINS - 1;
    for (int i = 0; i < NBINS; ++i) {
        if (off < offsets[i + 1]) { g = i; break; }
    }
    tileG[t] = g;
}

__global__ void k_scatter(const float* __restrict__ pseudo, const int* __restrict__ ei,
                          int* cursors, float2* pairSB, int* pairDst) {
    int e = blockIdx.x * blockDim.x + threadIdx.x;
    if (e >= EE) return;
    int src = ei[e], dst = ei[EE + e];
    float p0 = pseudo[e * 3 + 0], p1 = pseudo[e * 3 + 1], p2 = pseudo[e * 3 + 2];
#pragma unroll
    for (int s = 0; s < 8; ++s) {
        float bs; int g;
        spline_corner(p0, p1, p2, s, bs, g);
        int pos = atomicAdd(&cursors[g], 1);
        pairSB[pos] = make_float2(__int_as_float(src), bs);
        pairDst[pos] = dst;
    }
}

__global__ void k_zero(float* buf, int n) {
    int i = blockIdx.x * blockDim.x + threadIdx.x;
    if (i < n) buf[i] = 0.0f;
}

// ---- one-time swizzle of W (125,cin,cout) into WMMA-B lane order, 0-padded --
// Wlay[(g*4+kk)*32 + lane] = { W[g][k0][ncol], W[g][k0+1][ncol] },
//   k0 = kk*4 + (lane>>4)*2, ncol = lane&15
__global__ void k_padW(const float* __restrict__ W, int cin, int cout,
                       float2* __restrict__ Wlay) {
    int t = blockIdx.x * blockDim.x + threadIdx.x;
    if (t >= NBINS * 128) return;
    int g = t >> 7, r = t & 127;
    int kk = r >> 5, lane = r & 31;
    int hi = lane >> 4, ncol = lane & 15;
    int k0 = kk * 4 + hi * 2;
    const float* Wg = W + g * cin * cout;
    float2 v;
    v.x = (k0     < cin && ncol < cout) ? Wg[k0 * cout + ncol]       : 0.0f;
    v.y = (k0 + 1 < cin && ncol < cout) ? Wg[(k0 + 1) * cout + ncol] : 0.0f;
    Wlay[t] = v;
}

// same swizzle for root matrix R (cin,cout) + zero-padded bias (16)
__global__ void k_padR(const float* __restrict__ R, const float* __restrict__ bias,
                       int cin, int cout, float2* __restrict__ Rlay,
                       float* __restrict__ biasPad) {
    int t = blockIdx.x * blockDim.x + threadIdx.x;
    if (t < 128) {
        int kk = t >> 5, lane = t & 31;
        int hi = lane >> 4, ncol = lane & 15;
        int k0 = kk * 4 + hi * 2;
        float2 v;
        v.x = (k0     < cin && ncol < cout) ? R[k0 * cout + ncol]       : 0.0f;
        v.y = (k0 + 1 < cin && ncol < cout) ? R[(k0 + 1) * cout + ncol] : 0.0f;
        Rlay[t] = v;
    } else if (t < 144) {
        int n = t - 128;
        biasPad[n] = (n < cout) ? bias[n] : 0.0f;
    }
}

// ---------------- binned edge-message GEMM: wave = one 16-row tile ----------
// rows m: a[m,k] = basis * h[src[m]][k]  (k<16, zero-padded)
// B = pre-swizzled W[g]; D scatter-added to agg[dst] with f32 atomics.
__global__ void k_edge_wmma(const float* __restrict__ hin, const v2f* __restrict__ Wlay,
                            float* __restrict__ agg, const int* __restrict__ tileG,
                            const float2* __restrict__ pairSB,
                            const int* __restrict__ pairDst, int cout) {
    int wave = (blockIdx.x * blockDim.x + threadIdx.x) >> 5;
    if (wave >= NTILES) return;                 // wave-uniform exit (EXEC stays all-1s)
    int lane = threadIdx.x & 31;
    int off  = wave << 4;

    int g = tileG[wave];
    const v2f* Wg = Wlay + g * 128 + lane;

    int M  = lane & 15;       // A row for this lane; also N column of B/D
    int hi = lane >> 4;       // half-wave select (K offset +2 per ISA A layout)
    float2 sb = pairSB[off + M];
    int   srcn = __float_as_int(sb.x);
    float bas  = sb.y;
    const float* hrow = hin + (srcn * NCH + hi * 2);   // 32-bit index math

    v8f c = {};
#pragma unroll
    for (int kk = 0; kk < 4; ++kk) {
        v2f a = *(const v2f*)(hrow + kk * 4);   // b64 load of {h[k0], h[k0+1]}
        a.x *= bas; a.y *= bas;                 // v_pk_mul_f32
        v2f b = Wg[kk * 32];                    // b64 load, pre-swizzled/padded
        c = __builtin_amdgcn_wmma_f32_16x16x4_f32(false, a, false, b, (short)0, c,
                                                  false, false);
    }
    // D layout: VGPR r -> rows r (lanes 0-15) / r+8 (lanes 16-31), N = lane&15
    if (M < cout) {
#pragma unroll
        for (int r = 0; r < 8; ++r) {
            int idx = pairDst[off + r + 8 * hi] * NCH + M;   // 32-bit element index
            atomicAdd(agg + idx, c[r]);
        }
    }
}

// ---------------- node update: h@R (WMMA) + agg + bias, ELU, write ----------
// fully unguarded: R/bias/agg are zero-padded and ELU(0)==0.
__global__ void k_node_wmma(const float* __restrict__ hin, const float* __restrict__ agg,
                            const v2f* __restrict__ Rlay, const float* __restrict__ biasPad,
                            float* __restrict__ hout, float* __restrict__ out) {
    int wave = (blockIdx.x * blockDim.x + threadIdx.x) >> 5;
    int n0 = wave << 4;
    if (n0 >= NN) return;
    int lane = threadIdx.x & 31;
    int M = lane & 15, hi = lane >> 4;
    const float* arow = hin + ((n0 + M) * NCH + hi * 2);
    float bv = biasPad[M];

    v8f c = {};
#pragma unroll
    for (int kk = 0; kk < 4; ++kk) {
        v2f a = *(const v2f*)(arow + kk * 4);
        v2f b = Rlay[kk * 32 + lane];
        c = __builtin_amdgcn_wmma_f32_16x16x4_f32(false, a, false, b, (short)0, c,
                                                  false, false);
    }
#pragma unroll
    for (int r = 0; r < 8; ++r) {
        int node = n0 + r + 8 * hi;
        int idx  = node * NCH + M;
        float v = c[r] + agg[idx] + bv;
        float res = (v > 0.0f) ? v : (__expf(v) - 1.0f);   // ELU, alpha=1; ELU(0)=0
        hout[idx] = res;
        if (out != nullptr && M == 0) out[node] = res;     // final layer, cout==1
    }
}

// =====================================================================
extern "C" void kernel_launch(void* const* d_in, const int* in_sizes, int n_in,
                              void* d_out, int out_size, void* d_ws, size_t ws_size,
                              hipStream_t stream) {
    (void)in_sizes; (void)n_in; (void)out_size; (void)ws_size;
    const float* x      = (const float*)d_in[0];
    const int*   ei     = (const int*)  d_in[1];
    const float* pseudo = (const float*)d_in[2];
    const float* Wl[5]  = {(const float*)d_in[3],  (const float*)d_in[6],
                           (const float*)d_in[9],  (const float*)d_in[12],
                           (const float*)d_in[15]};
    const float* Rl[5]  = {(const float*)d_in[4],  (const float*)d_in[7],
                           (const float*)d_in[10], (const float*)d_in[13],
                           (const float*)d_in[16]};
    const float* bl[5]  = {(const float*)d_in[5],  (const float*)d_in[8],
                           (const float*)d_in[11], (const float*)d_in[14],
                           (const float*)d_in[17]};
    const int CH[6] = {1, 8, 16, 16, 8, 1};

    // workspace layout
    char*   ws      = (char*)d_ws;
    float*  red     = (float*)(ws + 0);
    int*    counts  = (int*)  (ws + 256);
    int*    offsets = (int*)  (ws + 1024);   // 126 entries
    int*    cursors = (int*)  (ws + 1792);
    float*  h0      = (float*)(ws + 4096);
    float*  h1      = h0  + (size_t)NN * NCH;
    float*  agg     = h1  + (size_t)NN * NCH;
    float2* pairSB  = (float2*)(agg + (size_t)NN * NCH);
    int*    pairDst = (int*)(pairSB + MCAP);
    int*    tileG   = pairDst + MCAP;
    float2* WlayAll = (float2*)(tileG + ((NTILES + 63) & ~63));  // 5 * 125*128 float2
    float2* RlayAll = WlayAll + (size_t)5 * NBINS * 128;         // 5 * 128 float2
    float*  bPadAll = (float*)(RlayAll + 5 * 128);               // 5 * 16 floats

    // ---- preprocessing (layer-invariant) ----
    k_init<<<(MCAP + 255) / 256, 256, 0, stream>>>(red, counts, h0, pairSB, pairDst);
    k_reduce<<<(NN + 255) / 256, 256, 0, stream>>>(x, red);
    k_normalize<<<(NN + 255) / 256, 256, 0, stream>>>(x, red, h0);
    k_count<<<(EE + 255) / 256, 256, 0, stream>>>(pseudo, counts);
    k_scan<<<1, 1, 0, stream>>>(counts, offsets, cursors);
    k_tileg<<<(NTILES + 255) / 256, 256, 0, stream>>>(offsets, tileG);
    k_scatter<<<(EE + 255) / 256, 256, 0, stream>>>(pseudo, ei, cursors, pairSB, pairDst);
    for (int l = 0; l < 5; ++l) {
        k_padW<<<(NBINS * 128 + 255) / 256, 256, 0, stream>>>(
            Wl[l], CH[l], CH[l + 1], WlayAll + (size_t)l * NBINS * 128);
        k_padR<<<1, 144, 0, stream>>>(Rl[l], bl[l], CH[l], CH[l + 1],
                                      RlayAll + l * 128, bPadAll + l * 16);
    }

    // ---- 5 spline-conv layers, ping-pong h buffers ----
    float* hbuf[2] = {h0, h1};
    int cur = 0;
    const int edgeBlocks = (NTILES + 7) / 8;     // 8 waves (tiles) per 256-thread block
    const int nodeBlocks = (NN / 16 + 7) / 8;
    for (int l = 0; l < 5; ++l) {
        k_zero<<<(NN * NCH + 255) / 256, 256, 0, stream>>>(agg, NN * NCH);
        k_edge_wmma<<<edgeBlocks, 256, 0, stream>>>(
            hbuf[cur], (const v2f*)(WlayAll + (size_t)l * NBINS * 128), agg, tileG,
            pairSB, pairDst, CH[l + 1]);
        k_node_wmma<<<nodeBlocks, 256, 0, stream>>>(
            hbuf[cur], agg, (const v2f*)(RlayAll + l * 128), bPadAll + l * 16,
            hbuf[1 - cur], (l == 4) ? (float*)d_out : nullptr);
        cur ^= 1;
    }
}